// ChannelPolyLayer_84971632984191
// MI455X (gfx1250) — compile-verified
//
#include <hip/hip_runtime.h>
#include <hip/hip_bf16.h>

// ChannelPolyLayer on MI455X (gfx1250, wave32).
//   img    : (16, 3, 512, 512) f32
//   coeffs : (16, 3, 20)       f32
//   powers : (20, 3)           f32   (fixed graded-lex exponent table -> hardcoded)
//   out    : (16, 3, 512, 512) f32
//
// Monomials on VALU (16 muls + 10 cndmask/pixel, all in named registers -> no
// LDS/scratch spill), the 3x20 x 20xN contraction on the matrix pipe via
// v_wmma_f32_16x16x4_f32 (5 chained K=4 chunks per 16-pixel tile).

typedef __attribute__((ext_vector_type(2))) float v2f;
typedef __attribute__((ext_vector_type(8))) float v8f;

#define CPL_HW        (512 * 512)
#define CPL_NOUT      3
#define CPL_NVARS     3
#define CPL_NC        20
#define CPL_TILES     4                 // 16-pixel tiles per wave
#define CPL_WAVES     8                 // waves per block (256 threads)
#define CPL_PIX_BLOCK (CPL_WAVES * CPL_TILES * 16)   // 512 pixels per block

__global__ __launch_bounds__(256)
void ChannelPolyLayer_wmma_kernel(const float* __restrict__ img,
                                  const float* __restrict__ coeffs,
                                  float* __restrict__ out)
{
    const int b    = blockIdx.y;
    const int lane = threadIdx.x & 31;
    const int wave = threadIdx.x >> 5;
    const bool hi  = lane >= 16;        // high half-wave supplies K=2,3 of each chunk
    const int  m   = lane & 15;         // A: output row M; B: pixel N within tile

    // ---- A matrix: coefficients, 5 chunks of K=4 (16x4 f32 -> 2 VGPRs each).
    // ISA 16x4 A layout: lanes 0-15 hold (M=lane, K0) in V0, (K1) in V1;
    // lanes 16-31 hold (M=lane-16, K2), (K3). Rows M>=3 are zero padding.
    const bool  valid = (m < CPL_NOUT);
    const int   mm    = valid ? m : 0;              // keep loads in-bounds
    const float* cp   = coeffs + (size_t)b * CPL_NOUT * CPL_NC
                               + (size_t)mm * CPL_NC + (hi ? 2 : 0);
    v2f a0 = *(const v2f*)(cp +  0);
    v2f a1 = *(const v2f*)(cp +  4);
    v2f a2 = *(const v2f*)(cp +  8);
    v2f a3 = *(const v2f*)(cp + 12);
    v2f a4 = *(const v2f*)(cp + 16);
    if (!valid) {                                   // zero padding rows (cndmask)
        a0 = (v2f)0.0f; a1 = (v2f)0.0f; a2 = (v2f)0.0f;
        a3 = (v2f)0.0f; a4 = (v2f)0.0f;
    }

    const size_t pixBase = (size_t)blockIdx.x * CPL_PIX_BLOCK
                         + (size_t)wave * (CPL_TILES * 16);
    const float* imgB = img + (size_t)b * CPL_NVARS * CPL_HW;
    float*       outB = out + (size_t)b * CPL_NOUT  * CPL_HW;

#pragma unroll
    for (int t = 0; t < CPL_TILES; ++t) {
        const size_t p = pixBase + (size_t)t * 16 + m;   // both halves: same 16 pixels
        const float x0 = imgB[p];
        const float x1 = imgB[CPL_HW + p];
        const float x2 = imgB[2 * CPL_HW + p];

        // 20 monomials as NAMED scalars (graded-lex, matches _generate_powers(3,3)):
        // 1, x0, x1, x2, x0^2, x0x1, x0x2, x1^2, x1x2, x2^2,
        // x0^3, x0^2x1, x0^2x2, x0x1^2, x0x1x2, x0x2^2, x1^3, x1^2x2, x1x2^2, x2^3
        const float m0  = 1.0f;
        const float m1  = x0;
        const float m2  = x1;
        const float m3  = x2;
        const float m4  = x0 * x0;
        const float m5  = x0 * x1;
        const float m6  = x0 * x2;
        const float m7  = x1 * x1;
        const float m8  = x1 * x2;
        const float m9  = x2 * x2;
        const float m10 = m4 * x0;
        const float m11 = m4 * x1;
        const float m12 = m4 * x2;
        const float m13 = m5 * x1;
        const float m14 = m5 * x2;
        const float m15 = m6 * x2;
        const float m16 = m7 * x1;
        const float m17 = m7 * x2;
        const float m18 = m8 * x2;
        const float m19 = m9 * x2;

        // B operands: lanes 0-15 supply rows K0 (V0) / K1 (V1); lanes 16-31 K2/K3.
        // Value-level ternaries -> v_cndmask_b32 in registers (no memory).
        v2f b0, b1, b2, b3, b4;
        b0.x = hi ? m2  : m0;   b0.y = hi ? m3  : m1;
        b1.x = hi ? m6  : m4;   b1.y = hi ? m7  : m5;
        b2.x = hi ? m10 : m8;   b2.y = hi ? m11 : m9;
        b3.x = hi ? m14 : m12;  b3.y = hi ? m15 : m13;
        b4.x = hi ? m18 : m16;  b4.y = hi ? m19 : m17;

        // ---- 5 chained WMMAs: D(16x16) += A(16x4) * B(4x16), K chunks of 4.
        v8f acc = {};
        acc = __builtin_amdgcn_wmma_f32_16x16x4_f32(false, a0, false, b0,
                                                    (short)0, acc, false, false);
        acc = __builtin_amdgcn_wmma_f32_16x16x4_f32(false, a1, false, b1,
                                                    (short)0, acc, false, false);
        acc = __builtin_amdgcn_wmma_f32_16x16x4_f32(false, a2, false, b2,
                                                    (short)0, acc, false, false);
        acc = __builtin_amdgcn_wmma_f32_16x16x4_f32(false, a3, false, b3,
                                                    (short)0, acc, false, false);
        acc = __builtin_amdgcn_wmma_f32_16x16x4_f32(false, a4, false, b4,
                                                    (short)0, acc, false, false);

        // D layout: VGPR r, lanes 0-15 => (M=r, N=lane). Rows 0..2 are the outputs.
        if (!hi) {
            outB[p]              = acc[0];   // o = 0
            outB[CPL_HW + p]     = acc[1];   // o = 1
            outB[2 * CPL_HW + p] = acc[2];   // o = 2
        }
    }
}

extern "C" void kernel_launch(void* const* d_in, const int* in_sizes, int n_in,
                              void* d_out, int out_size, void* d_ws, size_t ws_size,
                              hipStream_t stream) {
    (void)in_sizes; (void)n_in; (void)out_size; (void)d_ws; (void)ws_size;
    const float* img    = (const float*)d_in[0];
    const float* coeffs = (const float*)d_in[1];
    // d_in[2] (powers) is a fixed constant table; the monomial order is hardcoded.
    float* out = (float*)d_out;

    dim3 grid(CPL_HW / CPL_PIX_BLOCK, 16, 1);   // 512 x 16 blocks
    dim3 block(256, 1, 1);                      // 8 wave32 per block
    ChannelPolyLayer_wmma_kernel<<<grid, block, 0, stream>>>(img, coeffs, out);
}